// SIFLayerMaskModC2P_20581483282830
// MI455X (gfx1250) — compile-verified
//
#include <hip/hip_runtime.h>
#include <math.h>

#define POLAR_H 64
#define POLAR_W 512
#define IMG_B   64
#define IMG_H   480
#define IMG_W   640

typedef float vf4 __attribute__((ext_vector_type(4)));
typedef float vf2 __attribute__((ext_vector_type(2)));

// ---------------------------------------------------------------------------
// bicubic weights, torch coefficient A = -0.75, taps at offsets -1,0,1,2
// ---------------------------------------------------------------------------
__device__ __forceinline__ void cubic_w(float t, float w[4]) {
    const float A = -0.75f;
    float t1 = t + 1.0f;
    w[0] = ((A * t1 - 5.0f * A) * t1 + 8.0f * A) * t1 - 4.0f * A;
    w[1] = ((A + 2.0f) * t - (A + 3.0f)) * t * t + 1.0f;
    float u = 1.0f - t;
    w[2] = ((A + 2.0f) * u - (A + 3.0f)) * u * u + 1.0f;
    w[3] = 1.0f - w[0] - w[1] - w[2];
}

// bicubic sample at normalized coords (gx,gy), align_corners=true, zero pad.
// BIN: binarize loaded values against thresh before weighting (mask path).
// Fast path: fully interior sample -> 4 (possibly unaligned) b128 row loads.
template <bool BIN>
__device__ __forceinline__ float bicubic_sample(const float* __restrict__ base,
                                                float gx, float gy, float thresh) {
    float ix = (gx + 1.0f) * 0.5f * (float)(IMG_W - 1);
    float iy = (gy + 1.0f) * 0.5f * (float)(IMG_H - 1);
    float x0 = floorf(ix);
    float y0 = floorf(iy);
    float fx = ix - x0;
    float fy = iy - y0;
    int xi = (int)x0;
    int yi = (int)y0;
    float wx[4], wy[4];
    cubic_w(fx, wx);
    cubic_w(fy, wy);
    float acc = 0.0f;

    bool interior = (xi >= 1) && (xi + 2 <= IMG_W - 1) &&
                    (yi >= 1) && (yi + 2 <= IMG_H - 1);
    if (__builtin_expect(interior, 1)) {
#pragma unroll
        for (int j = 0; j < 4; ++j) {
            const float* row = base + (size_t)(yi + j - 1) * IMG_W + (xi - 1);
            vf4 v;
            __builtin_memcpy(&v, row, 16);   // dword-aligned b128 gather
            if (BIN) {
                v.x = (v.x > thresh) ? 1.0f : 0.0f;
                v.y = (v.y > thresh) ? 1.0f : 0.0f;
                v.z = (v.z > thresh) ? 1.0f : 0.0f;
                v.w = (v.w > thresh) ? 1.0f : 0.0f;
            }
            float rs = wx[0] * v.x + wx[1] * v.y + wx[2] * v.z + wx[3] * v.w;
            acc += wy[j] * rs;
        }
    } else {
        // border: exact masked scalar path (zero padding outside)
#pragma unroll
        for (int j = 0; j < 4; ++j) {
            int yy = yi + j - 1;
            bool vy = (yy >= 0) && (yy < IMG_H);
            int yc = min(max(yy, 0), IMG_H - 1);
            const float* row = base + (size_t)yc * IMG_W;
            float rs = 0.0f;
#pragma unroll
            for (int i = 0; i < 4; ++i) {
                int xx = xi + i - 1;
                bool vx = (xx >= 0) && (xx < IMG_W);
                int xc = min(max(xx, 0), IMG_W - 1);
                float v = (vy && vx) ? row[xc] : 0.0f;
                if (BIN) v = (v > thresh) ? 1.0f : 0.0f;
                rs += wx[i] * v;
            }
            acc += wy[j] * rs;
        }
    }
    return acc;
}

// gfx1250 global_prefetch_b8 of the 4 tap-rows of the mask gather
__device__ __forceinline__ void prefetch_taps(const float* __restrict__ base,
                                              float gx, float gy) {
    float ix = (gx + 1.0f) * 0.5f * (float)(IMG_W - 1);
    float iy = (gy + 1.0f) * 0.5f * (float)(IMG_H - 1);
    int xi = (int)floorf(ix) - 1;
    int yi = (int)floorf(iy) - 1;
    int xc = min(max(xi, 0), IMG_W - 1);
#pragma unroll
    for (int j = 0; j < 4; ++j) {
        int yc = min(max(yi + j, 0), IMG_H - 1);
        __builtin_prefetch(base + (size_t)yc * IMG_W + xc, 0, 3);
    }
}

// ---------------------------------------------------------------------------
// kernel 1: zero the max accumulator (d_ws is poisoned 0xAA by the harness)
// ---------------------------------------------------------------------------
__global__ void init_ws_kernel(unsigned* ws) {
    if (threadIdx.x == 0 && blockIdx.x == 0) ws[0] = 0u;
}

// ---------------------------------------------------------------------------
// kernel 2: exact global max of mask (all values >= 0 -> u32 bit-order max).
// b128 streaming loads (also warms the 192MB L2 with the mask for kernel 3),
// wave32 butterfly -> LDS -> global_atomic_max_u32.
// ---------------------------------------------------------------------------
__global__ void mask_max_kernel(const float* __restrict__ mask,
                                unsigned* __restrict__ ws, int n) {
    int gid = blockIdx.x * blockDim.x + threadIdx.x;
    int stride = gridDim.x * blockDim.x;
    int n4 = n >> 2;
    const vf4* __restrict__ m4 = (const vf4*)mask;
    float m = 0.0f;
    for (int i = gid; i < n4; i += stride) {
        vf4 v = m4[i];                         // global_load_b128
        m = fmaxf(m, fmaxf(fmaxf(v.x, v.y), fmaxf(v.z, v.w)));
    }
    if (gid == 0)                               // tail (n % 4), exact
        for (int i = n4 << 2; i < n; ++i) m = fmaxf(m, mask[i]);
#pragma unroll
    for (int off = 16; off > 0; off >>= 1)
        m = fmaxf(m, __shfl_xor(m, off, 32));
    __shared__ float sm[8];
    int lane = threadIdx.x & 31;
    int wv   = threadIdx.x >> 5;
    if (lane == 0) sm[wv] = m;
    __syncthreads();
    if (threadIdx.x == 0) {
        float bm = sm[0];
        int nw = blockDim.x >> 5;
        for (int i = 1; i < nw; ++i) bm = fmaxf(bm, sm[i]);
        atomicMax(ws, __float_as_uint(bm));
    }
}

// ---------------------------------------------------------------------------
// kernel 3: polar grid + bicubic sampling of image (with backprop offset) and
// binarized mask (no offset). One thread per (b, hp, wp).
// ---------------------------------------------------------------------------
__global__ void __launch_bounds__(256)
polar_kernel(const float* __restrict__ image,
             const float* __restrict__ mask,
             const float* __restrict__ pupil,     // [B,3]
             const float* __restrict__ iris,      // [B,3]
             const float* __restrict__ bpg,       // [B,Hp,Wp,2]
             const unsigned* __restrict__ maxbits,
             float* __restrict__ out, int total) {
    int tid = blockIdx.x * blockDim.x + threadIdx.x;
    if (tid >= total) return;
    int wp = tid & (POLAR_W - 1);
    int hp = (tid >> 9) & (POLAR_H - 1);
    int b  = tid >> 15;

    // ---- _polar_grid (same op sequence as reference) ----
    float theta = (6.28318530717958647692f * (float)wp) / (float)POLAR_W;
    float ct = cosf(theta);
    float st = sinf(theta);
    float pcx = pupil[b * 3 + 0], pcy = pupil[b * 3 + 1], pcr = pupil[b * 3 + 2];
    float icx = iris [b * 3 + 0], icy = iris [b * 3 + 1], icr = iris [b * 3 + 2];
    float px = pcx + pcr * ct, py = pcy + pcr * st;
    float qx = icx + icr * ct, qy = icy + icr * st;
    float r  = (float)(hp + 1) / (float)POLAR_H;
    float x  = (1.0f - r) * px + r * qx;
    float y  = (1.0f - r) * py + r * qy;
    float xn = (x - 1.0f) / (float)(IMG_W - 1) * 2.0f - 1.0f;
    float yn = (y - 1.0f) / (float)(IMG_H - 1) * 2.0f - 1.0f;

    // ---- _adjust_grid ----
    float gx = ((xn + 1.0f) / 2.0f * (float)IMG_W - 0.5f) / (float)(IMG_W - 1) * 2.0f - 1.0f;
    float gy = ((yn + 1.0f) / 2.0f * (float)IMG_H - 0.5f) / (float)(IMG_H - 1) * 2.0f - 1.0f;

    const float* ibase = image + (size_t)b * IMG_H * IMG_W;   // C == 1
    const float* mbase = mask  + (size_t)b * IMG_H * IMG_W;

    // prefetch the mask taps (gfx1250 global_prefetch_b8) so their lines are
    // in flight while the image-path ALU + gather executes
    prefetch_taps(mbase, gx, gy);

    // backprop_grid: 16MB read-once -> nontemporal b64 load (don't pollute L2)
    vf2 bp = __builtin_nontemporal_load((const vf2*)bpg + tid);

    // image path: grid + backprop offset; sampled on image*255 (linear -> scale after)
    float iv = bicubic_sample<false>(ibase, gx + bp.x, gy + bp.y, 0.0f) * 255.0f;

    // mask path: binarize (mask/max > 0.5  <=>  mask > 0.5*max), plain grid
    float maxv   = __uint_as_float(maxbits[0]);
    float thresh = 0.5f * maxv;
    float mv = bicubic_sample<true>(mbase, gx, gy, thresh);

    // outputs are write-once (16MB): NT stores keep image/mask L2-resident
    __builtin_nontemporal_store(iv, &out[tid]);          // image_polar
    __builtin_nontemporal_store(mv, &out[total + tid]);  // mask_polar
}

// ---------------------------------------------------------------------------
extern "C" void kernel_launch(void* const* d_in, const int* in_sizes, int n_in,
                              void* d_out, int out_size, void* d_ws, size_t ws_size,
                              hipStream_t stream) {
    const float* image = (const float*)d_in[0];
    const float* mask  = (const float*)d_in[1];
    const float* pupil = (const float*)d_in[2];
    const float* iris  = (const float*)d_in[3];
    const float* bpg   = (const float*)d_in[4];
    float* out = (float*)d_out;
    unsigned* ws = (unsigned*)d_ws;

    init_ws_kernel<<<1, 32, 0, stream>>>(ws);

    int n_mask = in_sizes[1];                  // 64*1*480*640
    mask_max_kernel<<<2048, 256, 0, stream>>>(mask, ws, n_mask);

    int total = IMG_B * POLAR_H * POLAR_W;     // 2,097,152
    polar_kernel<<<(total + 255) / 256, 256, 0, stream>>>(
        image, mask, pupil, iris, bpg, ws, out, total);
}